// GCN_16621523435856
// MI455X (gfx1250) — compile-verified
//
#include <hip/hip_runtime.h>
#include <hip/hip_bf16.h>

typedef __attribute__((ext_vector_type(2))) float v2f;
typedef __attribute__((ext_vector_type(8))) float v8f;

#define NUM_NODES   80000
#define NUM_EDGES   1280000
#define NUM_GRAPHS  512
#define HID         64
#define NUM_CLASSES 10

// ---------------------------------------------------------------------------
// Zero a float buffer
// ---------------------------------------------------------------------------
__global__ void zero_f32(float* __restrict__ p, int n) {
    int i = blockIdx.x * blockDim.x + threadIdx.x;
    if (i < n) p[i] = 0.0f;
}

// ---------------------------------------------------------------------------
// Edge scatter: agg[dst] += h[src]   (one wave32 per edge, 2 floats per lane)
// ---------------------------------------------------------------------------
__global__ void edge_scatter(const float* __restrict__ h,
                             const int* __restrict__ src,
                             const int* __restrict__ dst,
                             float* __restrict__ agg) {
    int wave = (int)((blockIdx.x * blockDim.x + threadIdx.x) >> 5);
    int lane = threadIdx.x & 31;
    if (wave >= NUM_EDGES) return;
    int s = src[wave];
    int d = dst[wave];
    v2f v = *(const v2f*)(h + (long)s * HID + lane * 2);
    atomicAdd(&agg[(long)d * HID + lane * 2 + 0], v.x);
    atomicAdd(&agg[(long)d * HID + lane * 2 + 1], v.y);
}

// ---------------------------------------------------------------------------
// GraphConv GEMM:  out = [relu]( agg @ w_rel^T + xin @ w_root^T + bias )
// One wave per 16x16 output tile; blockDim = (32,4) -> 4 col tiles (64 cols).
// Uses V_WMMA_F32_16X16X4_F32 (exact fp32), 32 WMMA per wave (K=64, 2 GEMMs).
//
// A layout (16x4 f32, ISA 7.12.2): lane m = lane%16, VGPR v: K = kk + 2*(lane/16) + v
// B = W^T  => B[k][n] = W[n][k]; same 2-consecutive-K-per-lane striping.
// C/D (16x16 f32): VGPR j: M = j + 8*(lane/16), N = lane%16.
// ---------------------------------------------------------------------------
__global__ void gconv_gemm(const float* __restrict__ agg,
                           const float* __restrict__ xin,
                           const float* __restrict__ w_rel,   // [HID][HID] row-major
                           const float* __restrict__ w_root,  // [HID][HID] row-major
                           const float* __restrict__ bias,    // [HID]
                           float* __restrict__ out,           // [N][HID]
                           int do_relu) {
    const int lane = threadIdx.x;          // 0..31
    const int ct   = threadIdx.y;          // col tile 0..3
    const int rt   = blockIdx.x;           // row tile 0..4999
    const int half = lane >> 4;            // 0 | 1
    const int l16  = lane & 15;

    const int row = rt * 16 + l16;         // node row for A operand
    const int col = ct * 16 + l16;         // output column for B/C/D

    const float* aggRow  = agg    + (long)row * HID;
    const float* xinRow  = xin    + (long)row * HID;
    const float* wrelRow = w_rel  + (long)col * HID;   // row n of W -> column n of B
    const float* wrtRow  = w_root + (long)col * HID;

    v8f acc = {};
#pragma unroll
    for (int kk = 0; kk < HID; kk += 4) {
        const int ka = kk + 2 * half;      // this lane's first K
        v2f a1 = *(const v2f*)(aggRow + ka);
        v2f b1 = *(const v2f*)(wrelRow + ka);
        v2f a2 = *(const v2f*)(xinRow + ka);
        v2f b2 = *(const v2f*)(wrtRow + ka);
        acc = __builtin_amdgcn_wmma_f32_16x16x4_f32(false, a1, false, b1,
                                                    (short)0, acc, false, false);
        acc = __builtin_amdgcn_wmma_f32_16x16x4_f32(false, a2, false, b2,
                                                    (short)0, acc, false, false);
    }

    const float bv = bias[col];
#pragma unroll
    for (int j = 0; j < 8; ++j) {
        const int m = j + 8 * half;
        float v = acc[j] + bv;
        if (do_relu) v = fmaxf(v, 0.0f);
        out[(long)(rt * 16 + m) * HID + col] = v;
    }
}

// ---------------------------------------------------------------------------
// Mean-pool accumulate: sums[batch[n]] += h[n]; counts[batch[n]] += 1
// One wave per node, 2 floats per lane.
// ---------------------------------------------------------------------------
__global__ void pool_accum(const float* __restrict__ h,
                           const int* __restrict__ batch,
                           float* __restrict__ sums,
                           float* __restrict__ counts) {
    int wave = (int)((blockIdx.x * blockDim.x + threadIdx.x) >> 5);
    int lane = threadIdx.x & 31;
    if (wave >= NUM_NODES) return;
    int g = batch[wave];
    v2f v = *(const v2f*)(h + (long)wave * HID + lane * 2);
    atomicAdd(&sums[(long)g * HID + lane * 2 + 0], v.x);
    atomicAdd(&sums[(long)g * HID + lane * 2 + 1], v.y);
    if (lane == 0) atomicAdd(&counts[g], 1.0f);
}

// ---------------------------------------------------------------------------
// Head: out[g][c] = (sums[g] . w_lin[c]) / max(count,1) + b_lin[c]
// (mean folded through the linear layer — linearity makes this exact)
// ---------------------------------------------------------------------------
__global__ void head_kernel(const float* __restrict__ sums,
                            const float* __restrict__ counts,
                            const float* __restrict__ w_lin,   // [10][64]
                            const float* __restrict__ b_lin,   // [10]
                            float* __restrict__ out) {         // [G][10]
    int g = blockIdx.x;
    int c = threadIdx.x;
    float inv = 1.0f / fmaxf(counts[g], 1.0f);
    if (c < NUM_CLASSES) {
        const float* srow = sums + (long)g * HID;
        const float* wrow = w_lin + (long)c * HID;
        float s = 0.0f;
#pragma unroll
        for (int k = 0; k < HID; ++k) s += srow[k] * wrow[k];
        out[g * NUM_CLASSES + c] = s * inv + b_lin[c];
    }
}

// ---------------------------------------------------------------------------
extern "C" void kernel_launch(void* const* d_in, const int* in_sizes, int n_in,
                              void* d_out, int out_size, void* d_ws, size_t ws_size,
                              hipStream_t stream) {
    const float* x        = (const float*)d_in[0];
    const int*   edge     = (const int*)d_in[1];     // [2][NUM_EDGES]
    const int*   batch    = (const int*)d_in[2];
    const float* w_rel1   = (const float*)d_in[3];
    const float* b_rel1   = (const float*)d_in[4];
    const float* w_root1  = (const float*)d_in[5];
    const float* w_rel2   = (const float*)d_in[6];
    const float* b_rel2   = (const float*)d_in[7];
    const float* w_root2  = (const float*)d_in[8];
    const float* w_rel3   = (const float*)d_in[9];
    const float* b_rel3   = (const float*)d_in[10];
    const float* w_root3  = (const float*)d_in[11];
    const float* w_lin    = (const float*)d_in[12];
    const float* b_lin    = (const float*)d_in[13];
    float*       out      = (float*)d_out;

    const int* src = edge;
    const int* dst = edge + NUM_EDGES;

    const size_t featN = (size_t)NUM_NODES * HID;    // 5,120,000 floats
    float* hA     = (float*)d_ws;                    // layer outputs (ping)
    float* hB     = hA + featN;                      // layer outputs (pong)
    float* agg    = hB + featN;                      // neighbor-sum scratch
    float* sums   = agg + featN;                     // [G][HID]
    float* counts = sums + (size_t)NUM_GRAPHS * HID; // [G]

    const int ZB = 256;
    const int zeroAggGrid  = (int)((featN + ZB - 1) / ZB);
    const int scatterGrid  = NUM_EDGES / 8;          // 8 waves / block, wave per edge
    const dim3 gemmBlock(32, 4);
    const int  gemmGrid = NUM_NODES / 16;            // 5000 row tiles
    const int  poolGrid = NUM_NODES / 8;

    // ---- layer 1: x -> hA (relu) ----
    zero_f32<<<zeroAggGrid, ZB, 0, stream>>>(agg, (int)featN);
    edge_scatter<<<scatterGrid, 256, 0, stream>>>(x, src, dst, agg);
    gconv_gemm<<<gemmGrid, gemmBlock, 0, stream>>>(agg, x, w_rel1, w_root1, b_rel1, hA, 1);

    // ---- layer 2: hA -> hB (relu) ----
    zero_f32<<<zeroAggGrid, ZB, 0, stream>>>(agg, (int)featN);
    edge_scatter<<<scatterGrid, 256, 0, stream>>>(hA, src, dst, agg);
    gconv_gemm<<<gemmGrid, gemmBlock, 0, stream>>>(agg, hA, w_rel2, w_root2, b_rel2, hB, 1);

    // ---- layer 3: hB -> hA (no relu) ----
    zero_f32<<<zeroAggGrid, ZB, 0, stream>>>(agg, (int)featN);
    edge_scatter<<<scatterGrid, 256, 0, stream>>>(hB, src, dst, agg);
    gconv_gemm<<<gemmGrid, gemmBlock, 0, stream>>>(agg, hB, w_rel3, w_root3, b_rel3, hA, 0);

    // ---- mean pool + head ----
    const int poolN = NUM_GRAPHS * HID + NUM_GRAPHS;   // sums + counts, contiguous
    zero_f32<<<(poolN + ZB - 1) / ZB, ZB, 0, stream>>>(sums, poolN);
    pool_accum<<<poolGrid, 256, 0, stream>>>(hA, batch, sums, counts);
    head_kernel<<<NUM_GRAPHS, 32, 0, stream>>>(sums, counts, w_lin, b_lin, out);
}